// DynamicConv2D_48773648613604
// MI455X (gfx1250) — compile-verified
//
#include <hip/hip_runtime.h>
#include <math.h>

typedef __attribute__((ext_vector_type(2))) float v2f;
typedef __attribute__((ext_vector_type(8))) float v8f;
typedef __attribute__((ext_vector_type(4))) unsigned int u32x4;
typedef __attribute__((ext_vector_type(8))) int i32x8;
typedef __attribute__((ext_vector_type(4))) int i32x4;

#define B_   32
#define C_   16
#define H_   256
#define W_   256
#define K_   4
#define O_   16
#define C4_  4
#define HW_  (H_ * W_)
#define KFULL 144            // C_ * 9 reduction length
#define AGG_PER_B 2304       // KFULL * O_
#define WPAD 264             // padded LDS row pitch (256 data + 8 pad DWORDs via TDM pad)

// ---- Tensor Data Mover: build D# (ISA ch.8) and issue tensor_load_to_lds.
// 2D tensor: dim0 (contiguous, data_size=4B), dim1 with stride0 elements.
// Optional LDS padding: pad_amt+1 DWORDs inserted every 2<<pad_int DWORDs.
__device__ __forceinline__ void tdm_load_2d(unsigned lds_addr, unsigned long long gaddr,
                                            unsigned dim0, unsigned dim1,
                                            unsigned stride0,
                                            unsigned tile0, unsigned tile1,
                                            unsigned pad_en, unsigned pad_int,
                                            unsigned pad_amt) {
    u32x4 g0;
    g0.x = 1u;                                           // count=1, user descriptor
    g0.y = lds_addr;                                     // LDS byte address
    g0.z = (unsigned)(gaddr & 0xffffffffull);            // global addr [31:0]
    g0.w = (unsigned)((gaddr >> 32) & 0x1ffffffull)      // global addr [56:32]
         | 0x80000000u;                                  // type=2 (bits 127:126 = 0b10)
    i32x8 g1;
    g1[0] = (int)((2u << 16)                             // data_size = 4B
                | (pad_en << 20)
                | (pad_int << 22)                        // 7 -> every 256 DWORDs
                | (pad_amt << 25));                      // 7 -> 8 DWORDs pad
    g1[1] = (int)((dim0 & 0xffffu) << 16);               // tensor_dim0[15:0]
    g1[2] = (int)(((dim0 >> 16) & 0xffffu) | ((dim1 & 0xffffu) << 16));
    g1[3] = (int)(((dim1 >> 16) & 0xffffu) | ((tile0 & 0xffffu) << 16));
    g1[4] = (int)(tile1 & 0xffffu);                      // tile_dim1 (tile_dim2=0)
    g1[5] = (int)stride0;                                // tensor_dim0_stride[31:0]
    g1[6] = 0;                                           // stride0[47:32]=0, stride1 lo=0
    g1[7] = 0;
    i32x4 z4 = {0, 0, 0, 0};
    i32x8 z8 = {0, 0, 0, 0, 0, 0, 0, 0};
    __builtin_amdgcn_tensor_load_to_lds(g0, g1, z4, z4, z8, 0);
}

// ---------------- Kernel 1: global average pool  x:(B,C,H,W) -> pooled:(B,C)
__global__ __launch_bounds__(256)
void pool_kernel(const float* __restrict__ x, float* __restrict__ pooled) {
    __shared__ float red[256];
    const int bc = blockIdx.x;                 // 0 .. B_*C_-1
    const float4* p4 = (const float4*)(x + (size_t)bc * HW_);
    float s = 0.f;
    for (int i = threadIdx.x; i < HW_ / 4; i += 256) {
        float4 v = p4[i];
        s += (v.x + v.y) + (v.z + v.w);
    }
    red[threadIdx.x] = s;
    __syncthreads();
    for (int off = 128; off > 0; off >>= 1) {
        if (threadIdx.x < off) red[threadIdx.x] += red[threadIdx.x + off];
        __syncthreads();
    }
    if (threadIdx.x == 0) pooled[bc] = red[0] * (1.0f / (float)HW_);
}

// ---------------- Kernel 2: attention MLP + softmax -> attn:(B,K)
__global__ __launch_bounds__(64)
void attn_kernel(const float* __restrict__ pooled,
                 const float* __restrict__ w1, const float* __restrict__ b1,
                 const float* __restrict__ w2, const float* __restrict__ b2,
                 float* __restrict__ attn) {
    const int b = threadIdx.x;
    if (b >= B_) return;
    float pc[C_];
    #pragma unroll
    for (int c = 0; c < C_; ++c) pc[c] = pooled[b * C_ + c];
    float h[C4_];
    #pragma unroll
    for (int j = 0; j < C4_; ++j) {
        float s = b1[j];
        #pragma unroll
        for (int c = 0; c < C_; ++c) s += pc[c] * w1[j * C_ + c];
        h[j] = fmaxf(s, 0.f);
    }
    float lg[K_];
    float mx = -1e30f;
    #pragma unroll
    for (int k = 0; k < K_; ++k) {
        float s = b2[k];
        #pragma unroll
        for (int j = 0; j < C4_; ++j) s += h[j] * w2[k * C4_ + j];
        lg[k] = s;
        mx = fmaxf(mx, s);
    }
    float den = 0.f;
    #pragma unroll
    for (int k = 0; k < K_; ++k) { lg[k] = __expf(lg[k] - mx); den += lg[k]; }
    const float inv = 1.f / den;
    #pragma unroll
    for (int k = 0; k < K_; ++k) attn[b * K_ + k] = lg[k] * inv;
}

// ---------------- Kernel 3: aggregate per-sample weights directly into WMMA
// A-fragment order: af[b][g][lane][j] with g = kfull>>2 (36 groups),
// lane = (half<<4)|o, j = kfull&1, so the conv kernel reads one v2f per K-step.
__global__ __launch_bounds__(256)
void agg_kernel(const float* __restrict__ weight, const float* __restrict__ attn,
                float* __restrict__ agg) {
    const int idx = blockIdx.x * 256 + threadIdx.x;
    if (idx >= B_ * AGG_PER_B) return;
    const int b   = idx / AGG_PER_B;
    const int rem = idx - b * AGG_PER_B;
    const int g    = rem >> 6;                 // K-group of 4
    const int lane = (rem >> 1) & 31;
    const int j    = rem & 1;
    const int o     = lane & 15;
    const int kfull = (g << 2) + ((lane >> 4) << 1) + j;
    const int c     = kfull & 15;
    const int k9    = kfull >> 4;              // 0..8 = ky*3+kx
    float s = 0.f;
    #pragma unroll
    for (int kk = 0; kk < K_; ++kk)
        s += attn[b * K_ + kk] * weight[(((size_t)kk * O_ + o) * C_ + c) * 9 + k9];
    agg[idx] = s;
}

// ---------------- Kernel 4: per-sample 3x3 conv as implicit GEMM, WMMA f32 16x16x4.
// One block per (b, output row h). TDM stages 3 halo rows x 16 ch + agg weights
// into LDS; 8 waves each run two interleaved 16-pixel accumulator chains.
__global__ __launch_bounds__(256)
void conv_wmma_kernel(const float* __restrict__ x, const float* __restrict__ agg,
                      float* __restrict__ out) {
    __shared__ __align__(16) float a_lds[AGG_PER_B];        // A fragments: 9216 B
    __shared__ __align__(16) float x_lds[3 * C_ * WPAD];    // [ky][c][col] : 50688 B

    const int blk = blockIdx.x;
    const int b   = blk >> 8;                   // sample
    const int h   = blk & 255;                  // output row
    const int tid = threadIdx.x;

    // Zero the halo columns TDM padding skips: col 0 and col 257 of all 48 rows.
    if (tid < 96) x_lds[(tid >> 1) * WPAD + ((tid & 1) ? 257 : 0)] = 0.f;

    // Zero whole row-regions for out-of-range image rows (h==0 / h==255 blocks).
    #pragma unroll
    for (int r = 0; r < 3; ++r) {
        const int hr = h + r - 1;
        if (hr < 0 || hr >= H_) {
            for (int i = tid; i < C_ * WPAD; i += 256)
                x_lds[r * C_ * WPAD + i] = 0.f;
        }
    }

    // Wave 0 issues the TDM descriptors; TENSORcnt waited before the barrier.
    if (tid < 32) {
        // Aggregated weights: 1D, 2304 contiguous floats, no padding.
        tdm_load_2d((unsigned)(uintptr_t)a_lds,
                    (unsigned long long)(uintptr_t)(agg + (size_t)b * AGG_PER_B),
                    AGG_PER_B, 1, AGG_PER_B, AGG_PER_B, 1, 0, 0, 0);
        // One 2D TDM per valid image row: dim0=W (row), dim1=C (stride H*W),
        // LDS pad 8 DWORDs per 256 -> pitch 264; +1 col offset for left halo.
        #pragma unroll
        for (int r = 0; r < 3; ++r) {
            const int hr = h + r - 1;
            if (hr >= 0 && hr < H_) {
                tdm_load_2d((unsigned)(uintptr_t)x_lds + (unsigned)(r * C_ * WPAD + 1) * 4u,
                            (unsigned long long)(uintptr_t)(x + ((size_t)b * C_ * H_ + hr) * W_),
                            W_, C_, HW_, W_, C_, 1, 7, 7);
            }
        }
        __builtin_amdgcn_s_wait_tensorcnt(0);
    }
    __syncthreads();

    const int wave = tid >> 5;
    const int lane = tid & 31;
    const int n    = lane & 15;                 // M index for A, N index for B/D
    const int half = lane >> 4;
    const int koff = half << 1;                 // K pair {0,1} vs {2,3}
    const int wb0  = wave << 4;                 // tile 0 pixel base
    const int wb1  = (wave + 8) << 4;           // tile 1 pixel base

    const v2f* __restrict__ afp = (const v2f*)a_lds;   // afp[g*32 + lane]

    v8f acc0 = {0.f, 0.f, 0.f, 0.f, 0.f, 0.f, 0.f, 0.f};
    v8f acc1 = {0.f, 0.f, 0.f, 0.f, 0.f, 0.f, 0.f, 0.f};

    #pragma unroll
    for (int k9 = 0; k9 < 9; ++k9) {
        const int ky = k9 / 3;
        const int kx = k9 - ky * 3;
        #pragma unroll
        for (int c0 = 0; c0 < 16; c0 += 4) {
            const int g  = k9 * 4 + (c0 >> 2);           // K-group index
            const v2f af = afp[g * 32 + lane];           // ds_load_b64 A fragment
            const int ci = c0 + koff;                    // channel for this half-wave
            const int ro = (ky * C_ + ci) * WPAD + n + kx;
            v2f bf0, bf1;
            bf0.x = x_lds[ro + wb0];
            bf0.y = x_lds[ro + wb0 + WPAD];              // channel ci+1
            bf1.x = x_lds[ro + wb1];
            bf1.y = x_lds[ro + wb1 + WPAD];
            // Two independent accumulator chains share the A fragment.
            acc0 = __builtin_amdgcn_wmma_f32_16x16x4_f32(
                       false, af, false, bf0, (short)0, acc0, false, false);
            acc1 = __builtin_amdgcn_wmma_f32_16x16x4_f32(
                       false, af, false, bf1, (short)0, acc1, false, false);
        }
    }

    // D 16x16 layout: VGPR v -> M = v + 8*half, N = lane&15
    #pragma unroll
    for (int v = 0; v < 8; ++v) {
        const int o = v + (half << 3);
        const size_t rowbase = (((size_t)b * O_ + o) * H_ + h) * W_;
        out[rowbase + wb0 + n] = acc0[v];
        out[rowbase + wb1 + n] = acc1[v];
    }
}

extern "C" void kernel_launch(void* const* d_in, const int* in_sizes, int n_in,
                              void* d_out, int out_size, void* d_ws, size_t ws_size,
                              hipStream_t stream) {
    const float* x      = (const float*)d_in[0];
    const float* weight = (const float*)d_in[1];
    const float* w1     = (const float*)d_in[2];
    const float* b1     = (const float*)d_in[3];
    const float* w2     = (const float*)d_in[4];
    const float* b2     = (const float*)d_in[5];
    float* out = (float*)d_out;

    float* ws     = (float*)d_ws;
    float* pooled = ws;            // 512 floats
    float* attn   = ws + 512;      // 128 floats
    float* agg    = ws + 640;      // 73728 floats (A-fragment order)

    pool_kernel<<<B_ * C_, 256, 0, stream>>>(x, pooled);
    attn_kernel<<<1, 64, 0, stream>>>(pooled, w1, b1, w2, b2, attn);
    agg_kernel<<<(B_ * AGG_PER_B + 255) / 256, 256, 0, stream>>>(weight, attn, agg);
    conv_wmma_kernel<<<B_ * H_, 256, 0, stream>>>(x, agg, out);
}